// SLODE_70265664963013
// MI455X (gfx1250) — compile-verified
//
#include <hip/hip_runtime.h>

// ---------------------------------------------------------------------------
// Blocked-scan LTI simulation on CDNA5 using V_WMMA_F32_16X16X4_F32 and
// GLOBAL_LOAD_ASYNC_TO_LDS_B128 for input staging.
//   nx=256, nu=ny=128, bsz=128, T=2048, chunk L=64, 32 chunks.
// Phase A: per-chunk carries (parallel).  A^64 by squaring.  Chunk-boundary
// scan (serial, 32 steps, WMMA).  Phase C: replay chunks + emit outputs.
// ---------------------------------------------------------------------------

typedef __attribute__((ext_vector_type(2))) float v2f;
typedef __attribute__((ext_vector_type(8))) float v8f;

#define NX 256
#define NU 128
#define NY 128
#define BSZ 128
#define TT 2048
#define CHUNK 64
#define NCHUNK (TT / CHUNK) /* 32 */
#define LDU 132              /* padded K-stride for U tiles in LDS */

// workspace offsets in floats
#define WS_E 0
#define WS_A (256 * 256)
#define WS_P0 (2 * 256 * 256)
#define WS_P1 (3 * 256 * 256)
#define WS_CARRY (4 * 256 * 256)
#define WS_XC (WS_CARRY + NCHUNK * NX * BSZ)

// Only meaningful in the device pass; host pass has no amdgcn builtins.
#if defined(__HIP_DEVICE_COMPILE__) && \
    !__has_builtin(__builtin_amdgcn_wmma_f32_16x16x4_f32)
#error "missing __builtin_amdgcn_wmma_f32_16x16x4_f32 on this toolchain"
#endif

__device__ __forceinline__ v8f wmma4(v2f a, v2f b, v8f c) {
#if defined(__HIP_DEVICE_COMPILE__)
  return __builtin_amdgcn_wmma_f32_16x16x4_f32(false, a, false, b, (short)0, c,
                                               false, false);
#else
  (void)a;
  (void)b;
  return c; // host stub; never executed
#endif
}

// byte offset of a shared-memory object within the LDS segment
__device__ __forceinline__ unsigned lds_off_u32(const void* p) {
#if defined(__HIP_DEVICE_COMPILE__)
  return (unsigned)(uintptr_t)(__attribute__((address_space(3))) const void*)p;
#else
  (void)p;
  return 0;
#endif
}

// async DMA: 16 bytes global -> LDS, tracked by ASYNCcnt (no VGPR round trip)
__device__ __forceinline__ void async_b128(unsigned lds_byte_off,
                                           const void* gaddr) {
#if defined(__HIP_DEVICE_COMPILE__)
  asm volatile("global_load_async_to_lds_b128 %0, %1, off"
               :
               : "v"(lds_byte_off), "v"((unsigned long long)(uintptr_t)gaddr)
               : "memory");
#else
  (void)lds_byte_off;
  (void)gaddr;
#endif
}

__device__ __forceinline__ void wait_async0() {
#if defined(__HIP_DEVICE_COMPILE__)
#if __has_builtin(__builtin_amdgcn_s_wait_asynccnt)
  __builtin_amdgcn_s_wait_asynccnt(0);
#else
  asm volatile("s_wait_asynccnt 0x0" ::: "memory");
#endif
#endif
}

// acc[i] = A[m0+16i : , :K] x Zl[:K, n0:n0+16]  (+acc), K = 4*KB.
// A row-major global (lda), Zl row-major LDS [K][LDN].
template <int NMT, int KB, int LDN>
__device__ __forceinline__ void gemm_tiles(const float* __restrict__ Amat,
                                           int lda, const float* Zl, int n0,
                                           int m0, v8f* acc, int l15, int hf) {
  for (int kb = 0; kb < KB; ++kb) {
    const int k0 = kb * 4 + hf * 2;
    v2f bf;
    bf.x = Zl[k0 * LDN + n0 + l15];
    bf.y = Zl[(k0 + 1) * LDN + n0 + l15];
#pragma unroll
    for (int i = 0; i < NMT; ++i) {
      const float2 av =
          *(const float2*)(Amat + (size_t)(m0 + i * 16 + l15) * lda + k0);
      v2f af;
      af.x = av.x;
      af.y = av.y;
      acc[i] = wmma4(af, bf, acc[i]);
    }
  }
}

// same, but B-operand from LDS in [N][LDUX] (K-contiguous) layout -> one b64
template <int NMT, int KB, int LDUX>
__device__ __forceinline__ void gemm_tiles_u(const float* __restrict__ Amat,
                                             int lda, const float* Ul, int n0,
                                             int m0, v8f* acc, int l15,
                                             int hf) {
  for (int kb = 0; kb < KB; ++kb) {
    const int k0 = kb * 4 + hf * 2;
    const float2 bv = *(const float2*)(Ul + (n0 + l15) * LDUX + k0);
    v2f bf;
    bf.x = bv.x;
    bf.y = bv.y;
#pragma unroll
    for (int i = 0; i < NMT; ++i) {
      const float2 av =
          *(const float2*)(Amat + (size_t)(m0 + i * 16 + l15) * lda + k0);
      v2f af;
      af.x = av.x;
      af.y = av.y;
      acc[i] = wmma4(af, bf, acc[i]);
    }
  }
}

// store a 16x16 f32 D fragment into LDS [rows][ldn] at (m0, n0)
__device__ __forceinline__ void store_acc_lds(float* Zl, int ldn, int m0,
                                              int n0, const v8f& acc, int l15,
                                              int hf) {
#pragma unroll
  for (int r = 0; r < 8; ++r)
    Zl[(m0 + r + hf * 8) * ldn + n0 + l15] = acc[r];
}

// --------------------------- small prep kernels ----------------------------

// E = 0.5*(S22 + S11), F = S21, S = M M^T + eps I (M is 512x512)
__global__ void __launch_bounds__(256) k_buildEF(const float* __restrict__ M,
                                                 float* __restrict__ ws) {
  const int i = blockIdx.x, j = threadIdx.x;
  const float* Mi = M + (size_t)i * 512;
  const float* Mfi = M + (size_t)(256 + i) * 512;
  const float* Mj = M + (size_t)j * 512;
  const float* Mfj = M + (size_t)(256 + j) * 512;
  float s_ij = 0.f, s_fij = 0.f, s_fifj = 0.f;
  for (int k = 0; k < 512; ++k) {
    const float mik = Mi[k], mfik = Mfi[k], mjk = Mj[k], mfjk = Mfj[k];
    s_ij += mik * mjk;
    s_fij += mfik * mjk;
    s_fifj += mfik * mfjk;
  }
  if (i == j) {
    s_ij += 1e-9f;
    s_fifj += 1e-9f;
  }
  ws[WS_E + i * 256 + j] = 0.5f * (s_fifj + s_ij);
  ws[WS_A + i * 256 + j] = s_fij; // F; becomes A in-place after k_gauss
}

// Gauss-Jordan (E SPD, no pivoting): in-place  ws[WS_A] <- E^{-1} F.
// Column-owner scheme: thread j owns column j of E and F.
__global__ void __launch_bounds__(256) k_gauss(float* __restrict__ ws) {
  float* E = ws + WS_E;
  float* A = ws + WS_A;
  const int j = threadIdx.x;
  __shared__ float colk[256];
  for (int k = 0; k < 256; ++k) {
    const float inv = 1.0f / E[k * 256 + k];
    const float ekj = E[k * 256 + j] * inv;
    const float akj = A[k * 256 + j] * inv;
    E[k * 256 + j] = ekj;
    A[k * 256 + j] = akj;
    colk[j] = (j == k) ? 0.0f : E[j * 256 + k];
    __syncthreads();
    for (int i = 0; i < 256; ++i) {
      const float f = colk[i];
      if (f != 0.0f) { // uniform branch (broadcast from LDS)
        E[i * 256 + j] -= f * ekj;
        A[i * 256 + j] -= f * akj;
      }
    }
    __threadfence_block();
    __syncthreads();
  }
}

// dst = a @ b, all 256x256 row-major (used for repeated squaring of A)
__global__ void __launch_bounds__(256) k_matmul256(float* __restrict__ dst,
                                                   const float* __restrict__ a,
                                                   const float* __restrict__ b) {
  const int i = blockIdx.x, j = threadIdx.x;
  float acc = 0.f;
  for (int k = 0; k < 256; ++k) acc += a[i * 256 + k] * b[k * 256 + j];
  dst[i * 256 + j] = acc;
}

// preds[:,0,:] = C x0 + D u[:,0,:]
__global__ void __launch_bounds__(128) k_y0(const float* __restrict__ u,
                                            const float* __restrict__ Cm,
                                            const float* __restrict__ Dm,
                                            const float* __restrict__ x0,
                                            float* __restrict__ preds) {
  const int b = blockIdx.x, j = threadIdx.x;
  float acc = 0.f;
  for (int n = 0; n < NX; ++n) acc += Cm[j * NX + n] * x0[n];
  const float* ub = u + (size_t)b * TT * NU;
  for (int k = 0; k < NU; ++k) acc += Dm[j * NU + k] * ub[k];
  preds[(size_t)b * TT * NY + j] = acc;
}

// ------------------------------ phase A ------------------------------------
// Chunk c, batch group bg (32 cols). Z=0; Z <- A Z + B u_t for L steps;
// write Z as carry_c.  Grid (NCHUNK-1, 4), block 256 (8 waves).
__global__ void __launch_bounds__(256) k_phaseA(const float* __restrict__ u,
                                                float* __restrict__ ws,
                                                const float* __restrict__ Bm) {
  __shared__ float Z[NX * 32];
  __shared__ float Ut[32 * LDU]; // [batch col][K], K-contiguous, padded
  const int c = blockIdx.x, bg = blockIdx.y, tid = threadIdx.x;
  const int b0 = bg * 32;
  const float* Am = ws + WS_A;
  for (int idx = tid; idx < NX * 32; idx += 256) Z[idx] = 0.0f;
  const int wave = tid >> 5, lane = tid & 31, l15 = lane & 15, hf = lane >> 4;
  const int n0 = (wave & 1) * 16;
  const int m0 = (wave >> 1) * 64; // 4 row tiles per wave
  const int jcol = tid >> 3, part = tid & 7;
  // per-thread staging addresses: 16 floats of u[b0+jcol, t, part*16 ..]
  const float* gbase0 =
      u + ((size_t)(b0 + jcol) * TT + (size_t)c * CHUNK) * NU + part * 16;
  const unsigned lbase = lds_off_u32(Ut) + (unsigned)(jcol * LDU + part * 16) * 4u;
  __syncthreads();
  for (int s = 0; s < CHUNK; ++s) {
    const float* gbase = gbase0 + (size_t)s * NU;
    // async DMA U_t (32 x 128) into LDS; prefetch next step's row into L2
#pragma unroll
    for (int q = 0; q < 4; ++q) async_b128(lbase + q * 16u, gbase + q * 4);
    if (s + 1 < CHUNK) __builtin_prefetch(gbase + NU, 0, 0);
    wait_async0();
    __syncthreads();
    v8f acc[4] = {};
    gemm_tiles<4, NX / 4, 32>(Am, NX, Z, n0, m0, acc, l15, hf);
    gemm_tiles_u<4, NU / 4, LDU>(Bm, NU, Ut, n0, m0, acc, l15, hf);
    __syncthreads();
#pragma unroll
    for (int i = 0; i < 4; ++i)
      store_acc_lds(Z, 32, m0 + i * 16, n0, acc[i], l15, hf);
    __syncthreads();
  }
  float* cr = ws + WS_CARRY + (size_t)c * NX * BSZ;
  for (int idx = tid; idx < NX * 32; idx += 256) {
    const int n = idx >> 5, jj = idx & 31;
    cr[n * BSZ + b0 + jj] = Z[idx];
  }
}

// ------------------------------ scan ---------------------------------------
// X_0 = x0 (broadcast); X_{c+1} = A^64 X_c + carry_c ; store every X_c.
// Grid 4 (batch groups of 32), block 256.
__global__ void __launch_bounds__(256) k_scan(float* __restrict__ ws,
                                              const float* __restrict__ x0) {
  __shared__ float Z[NX * 32];
  const int bg = blockIdx.x, tid = threadIdx.x;
  const int b0 = bg * 32;
  for (int idx = tid; idx < NX * 32; idx += 256) Z[idx] = x0[idx >> 5];
  const int wave = tid >> 5, lane = tid & 31, l15 = lane & 15, hf = lane >> 4;
  const int n0 = (wave & 1) * 16;
  const int m0 = (wave >> 1) * 64;
  const float* AL = ws + WS_P1; // A^64
  __syncthreads();
  for (int c = 0; c < NCHUNK; ++c) {
    float* xc = ws + WS_XC + (size_t)c * NX * BSZ;
    for (int idx = tid; idx < NX * 32; idx += 256) {
      const int n = idx >> 5, jj = idx & 31;
      xc[n * BSZ + b0 + jj] = Z[idx];
    }
    if (c == NCHUNK - 1) break;
    const float* cr = ws + WS_CARRY + (size_t)c * NX * BSZ;
    v8f acc[4];
#pragma unroll
    for (int i = 0; i < 4; ++i) {
#pragma unroll
      for (int r = 0; r < 8; ++r)
        acc[i][r] = cr[(m0 + i * 16 + r + hf * 8) * BSZ + b0 + n0 + l15];
    }
    gemm_tiles<4, NX / 4, 32>(AL, NX, Z, n0, m0, acc, l15, hf);
    __syncthreads();
#pragma unroll
    for (int i = 0; i < 4; ++i)
      store_acc_lds(Z, 32, m0 + i * 16, n0, acc[i], l15, hf);
    __syncthreads();
  }
}

// ------------------------------ phase C ------------------------------------
// Replay chunk from X_c; each step also Y = C Z_new + D U, written coalesced.
// Grid (NCHUNK, 4), block 256.
__global__ void __launch_bounds__(256) k_phaseC(
    const float* __restrict__ u, const float* __restrict__ ws,
    const float* __restrict__ Bm, const float* __restrict__ Cm,
    const float* __restrict__ Dm, float* __restrict__ preds) {
  __shared__ float Z[NX * 32];
  __shared__ float UY[32 * LDU]; // U tile ([N][K] padded); reused as Y^T stage
  const int c = blockIdx.x, bg = blockIdx.y, tid = threadIdx.x;
  const int b0 = bg * 32;
  const float* Am = ws + WS_A;
  const float* Xc = ws + WS_XC + (size_t)c * NX * BSZ;
  for (int idx = tid; idx < NX * 32; idx += 256) {
    const int n = idx >> 5, jj = idx & 31;
    Z[idx] = Xc[n * BSZ + b0 + jj];
  }
  const int wave = tid >> 5, lane = tid & 31, l15 = lane & 15, hf = lane >> 4;
  const int n0 = (wave & 1) * 16;
  const int m0 = (wave >> 1) * 64;  // state rows: 4 tiles/wave
  const int m0y = (wave >> 1) * 32; // output rows: 2 tiles/wave
  const int jcol = tid >> 3, part = tid & 7;
  const float* gbase0 =
      u + ((size_t)(b0 + jcol) * TT + (size_t)c * CHUNK) * NU + part * 16;
  const unsigned lbase = lds_off_u32(UY) + (unsigned)(jcol * LDU + part * 16) * 4u;
  __syncthreads();
  for (int s = 0; s < CHUNK; ++s) {
    const int t = c * CHUNK + s;
    if (t >= TT - 1) break; // last input u_{T-2} is the final step
    const float* gbase = gbase0 + (size_t)s * NU;
#pragma unroll
    for (int q = 0; q < 4; ++q) async_b128(lbase + q * 16u, gbase + q * 4);
    if (s + 1 < CHUNK && t + 1 < TT - 1) __builtin_prefetch(gbase + NU, 0, 0);
    wait_async0();
    __syncthreads();
    // Z_new = A Z + B U
    v8f acc[4] = {};
    gemm_tiles<4, NX / 4, 32>(Am, NX, Z, n0, m0, acc, l15, hf);
    gemm_tiles_u<4, NU / 4, LDU>(Bm, NU, UY, n0, m0, acc, l15, hf);
    __syncthreads();
#pragma unroll
    for (int i = 0; i < 4; ++i)
      store_acc_lds(Z, 32, m0 + i * 16, n0, acc[i], l15, hf);
    __syncthreads();
    // Y = C Z_new + D U
    v8f yacc[2] = {};
    gemm_tiles<2, NX / 4, 32>(Cm, NX, Z, n0, m0y, yacc, l15, hf);
    gemm_tiles_u<2, NU / 4, LDU>(Dm, NU, UY, n0, m0y, yacc, l15, hf);
    __syncthreads();
    // stage Y^T into (now free) UY as flat [ny][32]
#pragma unroll
    for (int i = 0; i < 2; ++i)
      store_acc_lds(UY, 32, m0y + i * 16, n0, yacc[i], l15, hf);
    __syncthreads();
    // coalesced write: preds[b][t+1][0:128], 512B contiguous per 8 threads
    float* op = preds + ((size_t)(b0 + jcol) * TT + (t + 1)) * NY + part * 16;
#pragma unroll
    for (int q = 0; q < 4; ++q) {
      float4 v;
      const int jb = part * 16 + q * 4;
      v.x = UY[(jb + 0) * 32 + jcol];
      v.y = UY[(jb + 1) * 32 + jcol];
      v.z = UY[(jb + 2) * 32 + jcol];
      v.w = UY[(jb + 3) * 32 + jcol];
      *(float4*)(op + q * 4) = v;
    }
    __syncthreads();
  }
}

// ------------------------------ launcher -----------------------------------

extern "C" void kernel_launch(void* const* d_in, const int* in_sizes, int n_in,
                              void* d_out, int out_size, void* d_ws,
                              size_t ws_size, hipStream_t stream) {
  (void)in_sizes;
  (void)n_in;
  (void)out_size;
  (void)ws_size;
  const float* u = (const float*)d_in[0];
  const float* M = (const float*)d_in[1];
  const float* B = (const float*)d_in[2];
  const float* C = (const float*)d_in[3];
  const float* D = (const float*)d_in[4];
  const float* x0 = (const float*)d_in[5];
  float* preds = (float*)d_out;
  float* ws = (float*)d_ws;

  // E, F from M; then A = E^{-1} F (in place at WS_A)
  k_buildEF<<<256, 256, 0, stream>>>(M, ws);
  k_gauss<<<1, 256, 0, stream>>>(ws);

  // per-chunk carries (parallel over 31 chunks x 4 batch groups)
  k_phaseA<<<dim3(NCHUNK - 1, 4), 256, 0, stream>>>(u, ws, B);

  // A^64 by repeated squaring -> WS_P1
  k_matmul256<<<256, 256, 0, stream>>>(ws + WS_P0, ws + WS_A, ws + WS_A);
  k_matmul256<<<256, 256, 0, stream>>>(ws + WS_P1, ws + WS_P0, ws + WS_P0);
  k_matmul256<<<256, 256, 0, stream>>>(ws + WS_P0, ws + WS_P1, ws + WS_P1);
  k_matmul256<<<256, 256, 0, stream>>>(ws + WS_P1, ws + WS_P0, ws + WS_P0);
  k_matmul256<<<256, 256, 0, stream>>>(ws + WS_P0, ws + WS_P1, ws + WS_P1);
  k_matmul256<<<256, 256, 0, stream>>>(ws + WS_P1, ws + WS_P0, ws + WS_P0);

  // chunk-boundary scan, then per-chunk replay emitting outputs
  k_scan<<<4, 256, 0, stream>>>(ws, x0);
  k_phaseC<<<dim3(NCHUNK, 4), 256, 0, stream>>>(u, ws, B, C, D, preds);
  k_y0<<<128, 128, 0, stream>>>(u, C, D, x0, preds);
}